// Loss_10660108828749
// MI455X (gfx1250) — compile-verified
//
#include <hip/hip_runtime.h>
#include <math.h>

// ---- problem constants (match reference) ----
#define Bsz   16
#define Tt    800
#define Ss    160
#define NMEL  80
#define Ll    4
#define Hh    8
#define Kk    5          // T / S
#define BW    16         // DC_BANDWIDTH
#define STOPW 8.0f
#define DCS   0.01f

#define NSLICE (Ll * Bsz * Hh)   // 512 (l, bh) slices
#define MEL_BLOCKS 256

typedef float v2f __attribute__((ext_vector_type(2)));
typedef float v8f __attribute__((ext_vector_type(8)));

__device__ __forceinline__ float waveReduceSum(float v) {
#pragma unroll
    for (int off = 16; off > 0; off >>= 1)
        v += __shfl_xor(v, off, 32);
    return v;
}

__device__ __forceinline__ float blockReduceSum(float v, float* smem) {
    const int lane = threadIdx.x & 31;
    const int wid  = threadIdx.x >> 5;
    v = waveReduceSum(v);
    if (lane == 0) smem[wid] = v;
    __syncthreads();
    const int nw = blockDim.x >> 5;
    float r = (threadIdx.x < nw) ? smem[threadIdx.x] : 0.0f;
    if (wid == 0) r = waveReduceSum(r);
    return r;  // valid in thread 0
}

// -------- kernel 1: masked L1 mel partial sums (float4 streaming) --------
__global__ void mel_partial_kernel(const float4* __restrict__ mp,
                                   const float4* __restrict__ mt,
                                   const int* __restrict__ length,
                                   float* __restrict__ ws_mel) {
    __shared__ int slen[Bsz];
    __shared__ float smem[8];
    if (threadIdx.x < Bsz) slen[threadIdx.x] = length[threadIdx.x];
    __syncthreads();

    const int n4 = Bsz * Tt * NMEL / 4;  // 256000, NMEL % 4 == 0 so (b,t) uniform per float4
    float acc = 0.0f;
    for (int i = blockIdx.x * blockDim.x + threadIdx.x; i < n4;
         i += gridDim.x * blockDim.x) {
        const int bt = (i * 4) / NMEL;        // b*T + t
        const int b  = bt / Tt;
        const int t  = bt - b * Tt;
        const float mk = (t < slen[b]) ? 1.0f : 0.0f;
        const float4 p = mp[i];
        const float4 q = mt[i];
        acc += fabsf(p.x * mk - q.x) + fabsf(p.y * mk - q.y) +
               fabsf(p.z * mk - q.z) + fabsf(p.w * mk - q.w);
    }
    const float s = blockReduceSum(acc, smem);
    if (threadIdx.x == 0) ws_mel[blockIdx.x] = s;
}

// -------- kernel 2: diagonal-constraint band sums via WMMA f32 16x16x4 --------
// One wave per (l, bh) slice. Band is zero for t >= 36, so only t-chunks
// t0 = 0..32 (step 4) are touched; s-chunks skipped when the band misses them
// (wave-uniform branch -> EXEC stays all-ones for WMMA).
__global__ void dc_partial_kernel(const float* __restrict__ al,
                                  const int* __restrict__ length,
                                  float* __restrict__ ws_dc) {
    const int lane  = threadIdx.x & 31;
    const int wave  = threadIdx.x >> 5;
    const int slice = blockIdx.x * (blockDim.x >> 5) + wave;  // 0..511
    if (slice >= NSLICE) return;

    const int bh = slice % (Bsz * Hh);
    const int b  = bh % Bsz;                 // reshape(L, H, B, ...) => bh = h*B + b
    const float* base = al + (size_t)slice * (size_t)(Ss * Tt);

    const int mrow  = lane & 15;             // A: M = lane%16 ; B: N = lane%16
    const int kbase = (lane >> 4) * 2;       // lanes 0-15 hold K={0,1}; 16-31 K={2,3}

    v8f c = {0.f, 0.f, 0.f, 0.f, 0.f, 0.f, 0.f, 0.f};

    for (int t0 = 0; t0 < 36; t0 += 4) {
        const int sLo = Kk * t0 - BW;               // union of band rows for this t-chunk
        const int sHi = Kk * (t0 + 3) + BW;         // exclusive
        for (int s0 = 0; s0 < Ss; s0 += 16) {
            if (s0 + 16 <= sLo || s0 >= sHi) continue;   // uniform skip

            const int srow = s0 + mrow;
            // A tile: al[srow, t0+kbase .. t0+kbase+1]  (8B-aligned: T even, offs even)
            const float2 av =
                *reinterpret_cast<const float2*>(base + (size_t)srow * Tt + (t0 + kbase));
            v2f a; a.x = av.x; a.y = av.y;

            // B tile (band^T): B[k,n] = band[s0+n, t0+k], n = lane%16
            const int tk0 = t0 + kbase, tk1 = t0 + kbase + 1;
            v2f bm;
            bm.x = (srow >= Kk * tk0 - BW && srow < Kk * tk0 + BW) ? 1.0f : 0.0f;
            bm.y = (srow >= Kk * tk1 - BW && srow < Kk * tk1 + BW) ? 1.0f : 0.0f;

            c = __builtin_amdgcn_wmma_f32_16x16x4_f32(
                    /*neg_a=*/false, a, /*neg_b=*/false, bm,
                    /*c_mod=*/(short)0, c, /*reuse_a=*/false, /*reuse_b=*/false);
        }
    }

    // diagonal of 16x16 f32 D: lanes 0-7 -> c[lane] (M=N=v), lanes 24-31 -> c[lane-24]
    float d = 0.0f;
#pragma unroll
    for (int v = 0; v < 8; ++v) {
        const float cv = c[v];
        d += ((lane == v) || (lane == 24 + v)) ? cv : 0.0f;
    }
    d = waveReduceSum(d);

    if (lane == 0) {
        const float sm = (Tt >= length[b]) ? 1.0f : 0.0f;  // smask
        ws_dc[slice] = d * sm;
    }
}

// -------- kernel 3: stop BCE + final combine --------
__global__ void final_kernel(const float* __restrict__ stop_pred,
                             const int* __restrict__ length,
                             const float* __restrict__ ws_mel,
                             const float* __restrict__ ws_dc,
                             float* __restrict__ out) {
    __shared__ float smem[8];

    float mv = (threadIdx.x < MEL_BLOCKS) ? ws_mel[threadIdx.x] : 0.0f;
    const float mel_sum = blockReduceSum(mv, smem);
    __syncthreads();

    const float dv = ws_dc[threadIdx.x] + ws_dc[threadIdx.x + 256];
    const float dc_sum = blockReduceSum(dv, smem);

    if (threadIdx.x == 0) {
        float bce = 0.0f;
        float lensum = 0.0f;
        for (int b = 0; b < Bsz; ++b) {
            const int len = length[b];
            lensum += (float)len;
            const float p = stop_pred[b * Tt + (len - 1)];   // last valid frame
            float lg = logf(p);
            if (lg < -100.0f) lg = -100.0f;                  // torch BCE clamp
            bce += -lg;
        }
        const float mel_loss  = mel_sum / (float)(Bsz * Tt * NMEL);
        const float stop_loss = STOPW * bce / lensum;
        const float dc        = dc_sum / ((float)Hh * lensum);
        out[0] = mel_loss + stop_loss - DCS * dc;
    }
}

extern "C" void kernel_launch(void* const* d_in, const int* in_sizes, int n_in,
                              void* d_out, int out_size, void* d_ws, size_t ws_size,
                              hipStream_t stream) {
    // setup_inputs order: length, mask, stop_pred, mels_pred, mels_target, alignments
    const int*   length      = (const int*)d_in[0];
    // d_in[1] = mask: recomputed from length (t < length[b]) -> not read
    const float* stop_pred   = (const float*)d_in[2];
    const float* mels_pred   = (const float*)d_in[3];
    const float* mels_target = (const float*)d_in[4];
    const float* alignments  = (const float*)d_in[5];

    float* ws      = (float*)d_ws;
    float* ws_mel  = ws;            // 256 floats
    float* ws_dc   = ws + 256;      // 512 floats

    mel_partial_kernel<<<MEL_BLOCKS, 256, 0, stream>>>(
        (const float4*)mels_pred, (const float4*)mels_target, length, ws_mel);

    dc_partial_kernel<<<NSLICE / 8, 256, 0, stream>>>(alignments, length, ws_dc);

    final_kernel<<<1, 256, 0, stream>>>(stop_pred, length, ws_mel, ws_dc,
                                        (float*)d_out);
}